// DetectionLoss_41455024341099
// MI455X (gfx1250) — compile-verified
//
#include <hip/hip_runtime.h>
#include <hip/hip_bf16.h>
#include <stdint.h>

// ---------------------------------------------------------------------------
// Detection loss for MI455X (gfx1250, wave32).
// Workspace record per (image,scale) entry e = b*3 + s, stride REC ints:
//   [0] box_l (f32)  [1] cls_l (f32)  [2] obj_pos (f32)
//   [3] npos (i32)   [4] ncells (i32)
//   [5] neg_masked_sum (f32, kernel2)  [6] neg_full_sum (f32, kernel2)
//   [8..71] exclusion cell indices (i32, up to 64)
// ---------------------------------------------------------------------------

#define REC 80
#define EPSF 1e-7f
#define IMGF 640.0f
#define FOUR_OVER_PI2 0.4052847345693511f

typedef __attribute__((ext_vector_type(2))) float v2f;
typedef __attribute__((ext_vector_type(8))) float v8f;

__device__ __forceinline__ float bce0(float x) {
    // bce_logits(x, 0) = max(x,0) + log1p(exp(-|x|))
    return fmaxf(x, 0.0f) + log1pf(expf(-fabsf(x)));
}

// -------------------------------- Kernel 1 ---------------------------------
// One block per image; threads 0..N-1 handle anchors for scale s.
__global__ void det_anchors_kernel(const float* __restrict__ bbox,
                                   const float* __restrict__ obj,
                                   const float* __restrict__ cls,
                                   const float* __restrict__ boxes,
                                   const int*   __restrict__ labels,
                                   int* __restrict__ ws,
                                   int B, int N, int C, int s, int H, int W,
                                   float min_sz, float max_sz, float stride_) {
    const int b   = blockIdx.x;
    const int tid = threadIdx.x;
    const int e   = b * 3 + s;
    int* rec = ws + (size_t)e * REC;

    __shared__ float s_box[64], s_cls[64], s_obj[64];
    __shared__ int   s_cells[64];
    __shared__ int   s_cnt;
    if (tid == 0) s_cnt = 0;
    __syncthreads();

    const int HW = H * W;
    float box_acc = 0.0f, cls_acc = 0.0f, objp_acc = 0.0f;

    for (int n = tid; n < N; n += blockDim.x) {
        const float* gt = boxes + ((size_t)b * N + n) * 4;
        const float cx = gt[0], cy = gt[1], w = gt[2], h = gt[3];
        const float tmax = fmaxf(w, h) * IMGF;
        const bool  mask = (tmax > min_sz) && (tmax <= max_sz);

        int gx = (int)(cx * (float)W); gx = min(max(gx, 0), W - 1);
        int gy = (int)(cy * (float)H); gy = min(max(gy, 0), H - 1);
        const int cell = gy * W + gx;

        // decode gathered bbox regression
        const float* rb = bbox + (size_t)b * 4 * HW;
        const float r0 = rb[cell], r1 = rb[HW + cell];
        const float r2 = rb[2 * HW + cell], r3 = rb[3 * HW + cell];
        const float inv = stride_ / IMGF;
        const float dcx = ((float)gx + 1.0f / (1.0f + expf(-r0))) * inv;
        const float dcy = ((float)gy + 1.0f / (1.0f + expf(-r1))) * inv;
        const float dw  = expf(r2) * inv;
        const float dh  = expf(r3) * inv;

        // CIoU(decoded, gt)
        const float px1 = dcx - dw * 0.5f, px2 = dcx + dw * 0.5f;
        const float py1 = dcy - dh * 0.5f, py2 = dcy + dh * 0.5f;
        const float qx1 = cx - w * 0.5f,  qx2 = cx + w * 0.5f;
        const float qy1 = cy - h * 0.5f,  qy2 = cy + h * 0.5f;
        const float iw = fmaxf(fminf(px2, qx2) - fmaxf(px1, qx1), 0.0f);
        const float ih = fmaxf(fminf(py2, qy2) - fmaxf(py1, qy1), 0.0f);
        const float inter = iw * ih;
        const float uni = (px2 - px1) * (py2 - py1) + (qx2 - qx1) * (qy2 - qy1) - inter;
        const float iou = inter / (uni + EPSF);
        const float cw = fmaxf(px2, qx2) - fminf(px1, qx1);
        const float ch = fmaxf(py2, qy2) - fminf(py1, qy1);
        const float c2 = cw * cw + ch * ch + EPSF;
        const float ddx = qx1 + qx2 - px1 - px2;
        const float ddy = qy1 + qy2 - py1 - py2;
        const float rho2 = (ddx * ddx + ddy * ddy) * 0.25f;
        const float wp = px2 - px1, hp = py2 - py1;
        const float wg = qx2 - qx1, hg = qy2 - qy1;
        const float da = atanf(wg / (hg + EPSF)) - atanf(wp / (hp + EPSF));
        const float v  = FOUR_OVER_PI2 * da * da;
        const float alpha = v / (1.0f - iou + v + EPSF);
        float ciou = iou - rho2 / c2 - alpha * v;
        ciou = fminf(fmaxf(ciou, -1.0f), 1.0f);

        const float m = mask ? 1.0f : 0.0f;
        box_acc += (1.0f - ciou) * m;

        // class BCE vs one-hot at gathered cell
        const int lab = labels[(size_t)b * N + n];
        const float* cb = cls + (size_t)b * C * HW + cell;
        float cl = 0.0f;
        for (int c = 0; c < C; ++c) {
            const float x = cb[(size_t)c * HW];
            const float t = (c == lab) ? 1.0f : 0.0f;
            cl += fmaxf(x, 0.0f) - x * t + log1pf(expf(-fabsf(x)));
        }
        cls_acc += cl * m;

        // positive objectness: bce(x, 1)
        const float xo = obj[(size_t)b * HW + cell];
        objp_acc += (fmaxf(xo, 0.0f) - xo + log1pf(expf(-fabsf(xo)))) * m;

        if (mask) {
            const int p = atomicAdd(&s_cnt, 1);
            if (p < 64) s_cells[p] = cell;
        }
    }

    s_box[tid] = box_acc; s_cls[tid] = cls_acc; s_obj[tid] = objp_acc;
    __syncthreads();

    if (tid == 0) {
        float tb = 0.0f, tc = 0.0f, to = 0.0f;
        for (int i = 0; i < 64; ++i) { tb += s_box[i]; tc += s_cls[i]; to += s_obj[i]; }
        ((float*)rec)[0] = tb;
        ((float*)rec)[1] = tc;
        ((float*)rec)[2] = to;
        rec[3] = s_cnt;                // npos == number of masked anchors
        rec[4] = min(s_cnt, 64);       // ncells stored
    }
    __syncthreads();
    const int nc = min(s_cnt, 64);
    if (tid < 64) rec[8 + tid] = (tid < nc) ? s_cells[tid] : -1;
}

// -------------------------------- Kernel 2 ---------------------------------
// One block per image; streams obj map through LDS via async global->LDS DMA
// (CDNA5 GLOBAL_LOAD_ASYNC_TO_LDS_B32 + s_wait_asynccnt), computes bce(x,0)
// and accumulates the mask_neg-weighted and full sums.
__global__ void det_objmap_kernel(const float* __restrict__ obj,
                                  int* __restrict__ ws,
                                  int s, int HW) {
    const int b   = blockIdx.x;
    const int tid = threadIdx.x;
    const int e   = b * 3 + s;
    int* rec = ws + (size_t)e * REC;

    __shared__ int   sc_cells[64];
    __shared__ int   sc_n;
    __shared__ float stage[256];
    __shared__ float wsum[8][2];

    if (tid == 0) sc_n = rec[4];
    if (tid < 64) sc_cells[tid] = rec[8 + tid];
    __syncthreads();
    const int ncells = sc_n;

    const float* base = obj + (size_t)b * HW;
    float full = 0.0f, masked = 0.0f;

    // LDS byte offset of this thread's staging slot (low 32 bits of the
    // generic shared-space address are the LDS offset on AMDGPU).
    const unsigned lds_addr = (unsigned)(uintptr_t)&stage[tid];

    const int ntiles = HW / 256;
    for (int t = 0; t < ntiles; ++t) {
        const int i = t * 256 + tid;
        const unsigned long long ga = (unsigned long long)(const void*)(base + i);
        asm volatile("global_load_async_to_lds_b32 %0, %1, off"
                     :: "v"(lds_addr), "v"(ga) : "memory");
        asm volatile("s_wait_asynccnt 0" ::: "memory");
        const float x = stage[tid];          // own slot only: no barrier needed
        const float v = bce0(x);
        full += v;
        bool ex = false;
        for (int j = 0; j < ncells; ++j) ex |= (sc_cells[j] == i);
        if (!ex) masked += v;
    }
    for (int i = ntiles * 256 + tid; i < HW; i += 256) {   // tail: plain loads
        const float v = bce0(base[i]);
        full += v;
        bool ex = false;
        for (int j = 0; j < ncells; ++j) ex |= (sc_cells[j] == i);
        if (!ex) masked += v;
    }

    // fixed-order wave32 tree reduce, then cross-wave via LDS
    for (int off = 16; off >= 1; off >>= 1) {
        full   += __shfl_xor(full,   off, 32);
        masked += __shfl_xor(masked, off, 32);
    }
    const int wid = tid >> 5, lane = tid & 31;
    if (lane == 0) { wsum[wid][0] = masked; wsum[wid][1] = full; }
    __syncthreads();
    if (tid == 0) {
        float m = 0.0f, f = 0.0f;
        for (int w = 0; w < 8; ++w) { m += wsum[w][0]; f += wsum[w][1]; }
        ((float*)rec)[5] = m;
        ((float*)rec)[6] = f;
    }
}

// -------------------------------- Kernel 3 ---------------------------------
// Single wave32. Per-entry combine, then cross-lane reduction through the
// matrix pipe: with A == all-ones, D[m][n] = sum_k B[k][n] independent of the
// exact B VGPR layout, so sum over one 16-lane half of d[0] equals the exact
// sum of all 64 B inputs (lane partial in VGPR0, 0 in VGPR1).
__device__ __forceinline__ float wmma_allreduce32(float p) {
    v2f a; a.x = 1.0f; a.y = 1.0f;         // A = ones (16x4)
    v2f bm; bm.x = p;  bm.y = 0.0f;        // B holds per-lane partials
    v8f c = {};
    c = __builtin_amdgcn_wmma_f32_16x16x4_f32(false, a, false, bm,
                                              (short)0, c, false, false);
    float v = c[0];
    // reduce within each 16-lane half; every lane ends with the full total
    v += __shfl_xor(v, 8, 32);
    v += __shfl_xor(v, 4, 32);
    v += __shfl_xor(v, 2, 32);
    v += __shfl_xor(v, 1, 32);
    return v;
}

__global__ void det_finalize_kernel(const int* __restrict__ ws,
                                    float* __restrict__ out, int B) {
    const int E = B * 3;
    const int lane = threadIdx.x;           // 32 threads, one wave
    float pbox = 0.0f, pobj = 0.0f, pcls = 0.0f, pnp = 0.0f;

    const int iters = (E + 31) / 32;
    for (int t = 0; t < iters; ++t) {
        const int e = t * 32 + lane;
        if (e < E) {
            const int*   rec  = ws + (size_t)e * REC;
            const float* recf = (const float*)rec;
            const float box  = recf[0], cls = recf[1], objp = recf[2];
            const int   npos = rec[3];
            const float negm = recf[5], negf = recf[6];
            const int   s    = e % 3;
            const float hw   = (s == 0) ? 6400.0f : (s == 1) ? 1600.0f : 400.0f;
            const bool  hp   = npos > 0;
            pbox += hp ? box : 0.0f;
            pcls += hp ? cls : 0.0f;
            pobj += hp ? (objp + 0.05f * negm) : (0.1f * negf / hw);
            pnp  += (float)npos;
        }
    }
    // EXEC is all-ones here (structured reconvergence) -> WMMA legal
    const float tb = wmma_allreduce32(pbox);
    const float to = wmma_allreduce32(pobj);
    const float tc = wmma_allreduce32(pcls);
    const float tn = wmma_allreduce32(pnp);

    if (lane == 0) {
        const float norm     = fmaxf(tn, 1.0f);
        const float box_loss = tb / norm;
        const float cls_loss = tc / norm;
        const float obj_loss = to / (float)(B * 3);
        out[0] = 7.5f * box_loss + 1.0f * obj_loss + 0.5f * cls_loss;
        out[1] = box_loss;
        out[2] = obj_loss;
        out[3] = cls_loss;
    }
}

// ------------------------------- launcher ----------------------------------
extern "C" void kernel_launch(void* const* d_in, const int* in_sizes, int n_in,
                              void* d_out, int out_size, void* d_ws, size_t ws_size,
                              hipStream_t stream) {
    const float* p3_bbox = (const float*)d_in[0];
    const float* p3_obj  = (const float*)d_in[1];
    const float* p3_cls  = (const float*)d_in[2];
    const float* p4_bbox = (const float*)d_in[3];
    const float* p4_obj  = (const float*)d_in[4];
    const float* p4_cls  = (const float*)d_in[5];
    const float* p5_bbox = (const float*)d_in[6];
    const float* p5_obj  = (const float*)d_in[7];
    const float* p5_cls  = (const float*)d_in[8];
    const float* boxes   = (const float*)d_in[9];
    const int*   labels  = (const int*)d_in[10];
    float* out = (float*)d_out;
    int*   ws  = (int*)d_ws;

    const int B = in_sizes[1] / (80 * 80);        // p3_obj = B*1*80*80
    const int N = in_sizes[10] / B;               // labels = B*N
    const int C = in_sizes[2] / (B * 80 * 80);    // p3_cls = B*C*80*80

    det_anchors_kernel<<<B, 64, 0, stream>>>(p3_bbox, p3_obj, p3_cls, boxes, labels,
                                             ws, B, N, C, 0, 80, 80, 0.0f, 64.0f, 8.0f);
    det_anchors_kernel<<<B, 64, 0, stream>>>(p4_bbox, p4_obj, p4_cls, boxes, labels,
                                             ws, B, N, C, 1, 40, 40, 64.0f, 128.0f, 16.0f);
    det_anchors_kernel<<<B, 64, 0, stream>>>(p5_bbox, p5_obj, p5_cls, boxes, labels,
                                             ws, B, N, C, 2, 20, 20, 128.0f, 10000.0f, 32.0f);

    det_objmap_kernel<<<B, 256, 0, stream>>>(p3_obj, ws, 0, 6400);
    det_objmap_kernel<<<B, 256, 0, stream>>>(p4_obj, ws, 1, 1600);
    det_objmap_kernel<<<B, 256, 0, stream>>>(p5_obj, ws, 2, 400);

    det_finalize_kernel<<<1, 32, 0, stream>>>(ws, out, B);
}